// ChamferHausdorffLoss_40475771797825
// MI455X (gfx1250) — compile-verified
//
#include <hip/hip_runtime.h>
#include <math.h>

typedef __attribute__((ext_vector_type(2))) float v2f;
typedef __attribute__((ext_vector_type(4))) float v4f;
typedef __attribute__((ext_vector_type(8))) float v8f;

#define NPTS 8192
#define BATCH 4

// Pack each target point j as the WMMA B-operand column:
// (-2gx, -2gy, -2gz, |g|^2). Lane-half 0 of the consumer reads .xy (K0,K1),
// lane-half 1 reads .zw (K2,K3) -- one b64 load per lane per tile.
__global__ __launch_bounds__(256)
void chamfer_pack(const float* __restrict__ G, float* __restrict__ pack) {
    const int idx = blockIdx.x * 256 + threadIdx.x;      // 0 .. BATCH*NPTS-1
    const float* g = G + (size_t)idx * 3;
    const float gx = g[0], gy = g[1], gz = g[2];
    v4f o;
    o.x = -2.0f * gx;
    o.y = -2.0f * gy;
    o.z = -2.0f * gz;
    o.w = gx * gx + gy * gy + gz * gz;
    *(v4f*)(pack + (size_t)idx * 4) = o;
}

// Each wave owns a 16-row tile of P and scans all packed targets in 16-column
// tiles. dist(i,j) = |p_i|^2 + ( -2 p_i . g_j + |g_j|^2 ); the bracketed term
// is one V_WMMA_F32_16X16X4_F32 per 16x16 tile with A = (px,py,pz,1).
// Row-min accumulates in the 8 D-layout VGPRs; |p|^2 is added after the min
// (per-row constant, doesn't change the argmin).
__global__ __launch_bounds__(256)
void chamfer_rowmin_wmma(const float* __restrict__ P,
                         const float* __restrict__ packG,
                         float* __restrict__ rowmin_out) {
    const int b = blockIdx.y;
    const float* Pb = P + (size_t)b * NPTS * 3;
    const float* Gp = packG + (size_t)b * NPTS * 4;
    float* outb = rowmin_out + (size_t)b * NPTS;

    const int lane = threadIdx.x & 31;
    const int wave = threadIdx.x >> 5;            // 0..7
    const int half = lane >> 4;                   // 0: K0/K1 slot, 1: K2/K3 slot
    const int lr   = lane & 15;                   // row/col index within tile
    const int rowBase = (blockIdx.x * 8 + wave) * 16;

    // Load this lane's pred point (lanes l and l+16 load the same row's point).
    const float* pp = Pb + (size_t)(rowBase + lr) * 3;
    const float px = pp[0], py = pp[1], pz = pp[2];
    const float pn2 = px * px + py * py + pz * pz;

    // A matrix (16x4 f32, 2 VGPRs): lanes 0-15 hold K=0,1; lanes 16-31 hold K=2,3.
    v2f A;
    A.x = half ? pz : px;     // K2 : K0
    A.y = half ? 1.0f : py;   // K3 : K1

    // Per-lane pointer into the packed B stream: point (tile*16 + lr), half word.
    const v2f* gp = (const v2f*)Gp + (size_t)lr * 2 + half;

    const v8f zeroC = {};
    v8f acc;
#pragma unroll
    for (int v = 0; v < 8; ++v) acc[v] = __builtin_inff();

#pragma unroll 4
    for (int t = 0; t < NPTS / 16; ++t) {
        const v2f B = gp[(size_t)t * 32];          // tile stride = 16 pts * 2 v2f
        v8f d = __builtin_amdgcn_wmma_f32_16x16x4_f32(
            /*neg_a=*/false, A, /*neg_b=*/false, B,
            /*c_mod=*/(short)0, zeroC, /*reuse_a=*/false, /*reuse_b=*/false);
#pragma unroll
        for (int v = 0; v < 8; ++v) acc[v] = fminf(acc[v], d[v]);
    }

    // acc[v] on lane l = min over all tiles of D[row = v + 8*half][col class lr].
    // Finish each row's min across its 16 column classes via LDS.
    __shared__ float red[8][16][16];               // [wave][row][colclass]
#pragma unroll
    for (int v = 0; v < 8; ++v) red[wave][v + 8 * half][lr] = acc[v];
    __syncthreads();

    if (lane < 16) {
        float mn = red[wave][lane][0];
#pragma unroll
        for (int c = 1; c < 16; ++c) mn = fminf(mn, red[wave][lane][c]);
        outb[rowBase + lane] = mn + pn2;           // add |p_i|^2 after the min
    }
}

// Reduce each 8192-element segment to (sum, max). 8 segments:
// 0..3 = min1 per batch, 4..7 = min2 per batch (contiguous in ws).
__global__ __launch_bounds__(256)
void chamfer_seg_reduce(const float* __restrict__ in, float* __restrict__ partials) {
    __shared__ float ssum[256];
    __shared__ float smax[256];
    const int seg = blockIdx.x;
    const float* p = in + (size_t)seg * NPTS;

    float s = 0.0f, mx = -__builtin_inff();
    for (int i = threadIdx.x; i < NPTS; i += 256) {
        const float v = p[i];
        s += v;
        mx = fmaxf(mx, v);
    }
    ssum[threadIdx.x] = s;
    smax[threadIdx.x] = mx;
    __syncthreads();
    for (int off = 128; off > 0; off >>= 1) {
        if (threadIdx.x < off) {
            ssum[threadIdx.x] += ssum[threadIdx.x + off];
            smax[threadIdx.x] = fmaxf(smax[threadIdx.x], smax[threadIdx.x + off]);
        }
        __syncthreads();
    }
    if (threadIdx.x == 0) {
        partials[2 * seg]     = ssum[0];
        partials[2 * seg + 1] = smax[0];
    }
}

// chamfer = mean_b( mean(min1_b) + mean(min2_b) )
// hausdorff = mean_b( max(min1_b) ) + mean_b( max(min2_b) )
__global__ void chamfer_finalize(const float* __restrict__ partials, float* __restrict__ out) {
    if (threadIdx.x == 0 && blockIdx.x == 0) {
        float ch = 0.0f, h1 = 0.0f, h2 = 0.0f;
        for (int b = 0; b < BATCH; ++b) {
            ch += partials[2 * b] * (1.0f / NPTS) + partials[2 * (b + BATCH)] * (1.0f / NPTS);
            h1 += partials[2 * b + 1];
            h2 += partials[2 * (b + BATCH) + 1];
        }
        out[0] = ch * (1.0f / BATCH);
        out[1] = (h1 + h2) * (1.0f / BATCH);
    }
}

extern "C" void kernel_launch(void* const* d_in, const int* in_sizes, int n_in,
                              void* d_out, int out_size, void* d_ws, size_t ws_size,
                              hipStream_t stream) {
    const float* pred = (const float*)d_in[0];
    const float* gt   = (const float*)d_in[1];
    float* out = (float*)d_out;

    float* min1     = (float*)d_ws;             // [4][8192]
    float* min2     = min1 + BATCH * NPTS;      // [4][8192]
    float* partials = min2 + BATCH * NPTS;      // [8][2] (sum, max)
    float* pack     = partials + 16;            // [4][8192][4], reused per direction

    dim3 block(256);
    dim3 packGrid(BATCH * NPTS / 256);          // 128 blocks
    dim3 grid(NPTS / 16 / 8, BATCH);            // (64, 4): one 16-row tile per wave

    // Direction 1: rows = pred, targets = gt
    chamfer_pack<<<packGrid, block, 0, stream>>>(gt, pack);
    chamfer_rowmin_wmma<<<grid, block, 0, stream>>>(pred, pack, min1);
    // Direction 2: rows = gt, targets = pred (pack buffer reused sequentially)
    chamfer_pack<<<packGrid, block, 0, stream>>>(pred, pack);
    chamfer_rowmin_wmma<<<grid, block, 0, stream>>>(gt, pack, min2);

    chamfer_seg_reduce<<<dim3(2 * BATCH), block, 0, stream>>>(min1, partials);
    chamfer_finalize<<<dim3(1), dim3(32), 0, stream>>>(partials, out);
}